// GAT_90744069030483
// MI455X (gfx1250) — compile-verified
//
#include <hip/hip_runtime.h>

#define N_NODES   50000
#define N_EDGES   800000
#define E_TOT     (N_EDGES + N_NODES)   // 850000 with self loops
#define N_GRAPHS  64
#define OUT_C     64
#define NEG_SLOPE 0.2f
#define HSZ       (N_NODES * 128)

typedef float v2f __attribute__((ext_vector_type(2)));
typedef float v8f __attribute__((ext_vector_type(8)));

// ---------------------------------------------------------------------------
// H = X @ W  using V_WMMA_F32_16X16X4_F32 (fp32 in/out, one wave per 16x16 tile)
// A (16x4 f32): lanes 0-15 hold M=0..15 {K=k,k+1}; lanes 16-31 hold {K=k+2,k+3}
// B (4x16 f32): VGPR0 = row K (lanes 0-15) / K+2 (lanes 16-31); VGPR1 = K+1/K+3
// D (16x16 f32): VGPR r -> row (half*8 + r), col = lane&15
// ---------------------------------------------------------------------------
__global__ __launch_bounds__(256)
void gat_gemm_wmma(const float* __restrict__ X, const float* __restrict__ W,
                   float* __restrict__ H, int din, int dout)
{
    int wave = blockIdx.x * (blockDim.x >> 5) + (threadIdx.x >> 5);
    int lane = threadIdx.x & 31;
    int nTilesN = dout >> 4;
    int nTiles  = (N_NODES >> 4) * nTilesN;
    if (wave >= nTiles) return;                 // wave-uniform: EXEC stays all-1s
    int mt = wave / nTilesN, nt = wave % nTilesN;
    int m0 = mt << 4, n0 = nt << 4;
    int half = lane >> 4;
    int lrow = lane & 15;

    const float* arow = X + (size_t)(m0 + lrow) * din;
    v8f acc = {};
    for (int k = 0; k < din; k += 4) {
        int ka = k + (half << 1);
        v2f a, b;
        a.x = arow[ka];
        a.y = arow[ka + 1];
        b.x = W[(size_t)ka * dout + n0 + lrow];
        b.y = W[(size_t)(ka + 1) * dout + n0 + lrow];
        acc = __builtin_amdgcn_wmma_f32_16x16x4_f32(
            /*neg_a=*/false, a, /*neg_b=*/false, b,
            /*c_mod=*/(short)0, acc, /*reuse_a=*/false, /*reuse_b=*/false);
    }
    float* orow = H + (size_t)(m0 + half * 8) * dout + n0 + lrow;
#pragma unroll
    for (int r = 0; r < 8; ++r)
        orow[(size_t)r * dout] = acc[r];
}

// alpha_src[n] = H[n]·a_src ; alpha_dst[n] = H[n]·a_dst (one wave per node)
__global__ __launch_bounds__(256)
void gat_alpha(const float* __restrict__ H, const float* __restrict__ avs,
               const float* __restrict__ avd, float* __restrict__ as_,
               float* __restrict__ ad_, int dout)
{
    int node = blockIdx.x * (blockDim.x >> 5) + (threadIdx.x >> 5);
    int lane = threadIdx.x & 31;
    if (node >= N_NODES) return;
    const float* h = H + (size_t)node * dout;
    float s = 0.f, d = 0.f;
    for (int c = lane; c < dout; c += 32) {
        float hv = h[c];
        s += hv * avs[c];
        d += hv * avd[c];
    }
#pragma unroll
    for (int off = 16; off > 0; off >>= 1) {
        s += __shfl_down(s, off, 32);
        d += __shfl_down(d, off, 32);
    }
    if (lane == 0) { as_[node] = s; ad_[node] = d; }
}

__global__ void fill_f32(float* __restrict__ p, float v, int n)
{
    int i = blockIdx.x * blockDim.x + threadIdx.x;
    if (i < n) p[i] = v;
}

__device__ __forceinline__ void edge_nodes(const int* __restrict__ ei, int e,
                                           int& s, int& d)
{
    if (e < N_EDGES) { s = ei[e]; d = ei[N_EDGES + e]; }
    else             { s = e - N_EDGES; d = s; }       // self loop
}

__device__ __forceinline__ void atomicMaxF(float* addr, float val)
{
    int* ai = (int*)addr;
    int old = *ai;
    while (__int_as_float(old) < val) {
        int assumed = old;
        old = atomicCAS(ai, assumed, __float_as_int(val));
        if (old == assumed) break;
    }
}

// pass 1: logits + segment max
__global__ void edge_max(const int* __restrict__ ei, const float* __restrict__ as_,
                         const float* __restrict__ ad_, float* __restrict__ mmax,
                         float* __restrict__ elog)
{
    int e = blockIdx.x * blockDim.x + threadIdx.x;
    if (e >= E_TOT) return;
    int s, d; edge_nodes(ei, e, s, d);
    float l = as_[s] + ad_[d];
    l = (l > 0.f) ? l : NEG_SLOPE * l;
    elog[e] = l;
    atomicMaxF(&mmax[d], l);
}

// pass 2: exp + segment sum (elog overwritten with exp values)
__global__ void edge_exp(const int* __restrict__ ei, float* __restrict__ elog,
                         const float* __restrict__ mmax, float* __restrict__ denom)
{
    int e = blockIdx.x * blockDim.x + threadIdx.x;
    if (e >= E_TOT) return;
    int s, d; edge_nodes(ei, e, s, d);
    float ev = __expf(elog[e] - mmax[d]);
    elog[e] = ev;
    atomicAdd(&denom[d], ev);
}

// pass 3: OUT[dst] += (e/denom[dst]) * H[src]   (one wave per edge, coalesced)
__global__ __launch_bounds__(256)
void edge_agg(const int* __restrict__ ei, const float* __restrict__ elog,
              const float* __restrict__ denom, const float* __restrict__ H,
              float* __restrict__ OUT, int dout)
{
    int e = blockIdx.x * (blockDim.x >> 5) + (threadIdx.x >> 5);
    int lane = threadIdx.x & 31;
    if (e >= E_TOT) return;
    int s, d; edge_nodes(ei, e, s, d);
    float w = elog[e] / denom[d];
    const float* hs = H + (size_t)s * dout;
    float* od = OUT + (size_t)d * dout;
    for (int c = lane; c < dout; c += 32)
        atomicAdd(&od[c], w * hs[c]);
}

__global__ void bias_act(float* __restrict__ OUT, const float* __restrict__ b,
                         int dout, int relu)
{
    int i = blockIdx.x * blockDim.x + threadIdx.x;
    if (i >= N_NODES * dout) return;
    float v = OUT[i] + b[i % dout];
    if (relu) v = fmaxf(v, 0.f);
    OUT[i] = v;
}

// global_add_pool into [N_GRAPHS, OUT_C]
__global__ void pool_kernel(const float* __restrict__ H, const int* __restrict__ batch,
                            float* __restrict__ out)
{
    int i = blockIdx.x * blockDim.x + threadIdx.x;
    if (i >= N_NODES * OUT_C) return;
    int n = i >> 6, c = i & 63;
    atomicAdd(&out[batch[n] * OUT_C + c], H[i]);
}

// ---------------------------------------------------------------------------
static void run_layer(const float* Xin, const float* W, const float* a_s,
                      const float* a_d, const float* bias, int din, int dout,
                      int relu, float* Hbuf, float* Obuf, float* as_, float* ad_,
                      float* mm, float* dn, float* el, const int* ei,
                      hipStream_t stream)
{
    int tiles = (N_NODES / 16) * (dout / 16);
    gat_gemm_wmma<<<(tiles + 7) / 8, 256, 0, stream>>>(Xin, W, Hbuf, din, dout);
    gat_alpha<<<(N_NODES + 7) / 8, 256, 0, stream>>>(Hbuf, a_s, a_d, as_, ad_, dout);
    fill_f32<<<(N_NODES + 255) / 256, 256, 0, stream>>>(mm, -3.0e38f, N_NODES);
    fill_f32<<<(N_NODES + 255) / 256, 256, 0, stream>>>(dn, 0.f, N_NODES);
    fill_f32<<<(N_NODES * dout + 255) / 256, 256, 0, stream>>>(Obuf, 0.f, N_NODES * dout);
    edge_max<<<(E_TOT + 255) / 256, 256, 0, stream>>>(ei, as_, ad_, mm, el);
    edge_exp<<<(E_TOT + 255) / 256, 256, 0, stream>>>(ei, el, mm, dn);
    edge_agg<<<(E_TOT + 7) / 8, 256, 0, stream>>>(ei, el, dn, Hbuf, Obuf, dout);
    bias_act<<<(N_NODES * dout + 255) / 256, 256, 0, stream>>>(Obuf, bias, dout, relu);
}

extern "C" void kernel_launch(void* const* d_in, const int* in_sizes, int n_in,
                              void* d_out, int out_size, void* d_ws, size_t ws_size,
                              hipStream_t stream)
{
    const float* x     = (const float*)d_in[0];
    const int*   ei    = (const int*)d_in[1];
    const int*   batch = (const int*)d_in[2];
    const float* W0 = (const float*)d_in[3];
    const float* s0 = (const float*)d_in[4];
    const float* d0 = (const float*)d_in[5];
    const float* b0 = (const float*)d_in[6];
    const float* W1 = (const float*)d_in[7];
    const float* s1 = (const float*)d_in[8];
    const float* d1 = (const float*)d_in[9];
    const float* b1 = (const float*)d_in[10];
    const float* W2 = (const float*)d_in[11];
    const float* s2 = (const float*)d_in[12];
    const float* d2 = (const float*)d_in[13];
    const float* b2 = (const float*)d_in[14];

    float* ws  = (float*)d_ws;
    float* hA  = ws;                    // N_NODES*128
    float* hB  = hA + HSZ;              // N_NODES*128
    float* as_ = hB + HSZ;              // N_NODES
    float* ad_ = as_ + N_NODES;         // N_NODES
    float* mm  = ad_ + N_NODES;         // N_NODES
    float* dn  = mm + N_NODES;          // N_NODES
    float* el  = dn + N_NODES;          // E_TOT

    // layer 0: x -> hA (gemm) -> hB (aggregated, relu)
    run_layer(x,  W0, s0, d0, b0, 128, 128, 1, hA, hB, as_, ad_, mm, dn, el, ei, stream);
    // layer 1: hB -> hA (gemm) -> hB
    run_layer(hB, W1, s1, d1, b1, 128, 128, 1, hA, hB, as_, ad_, mm, dn, el, ei, stream);
    // layer 2: hB -> hA (gemm, dout=64) -> hB (no relu)
    run_layer(hB, W2, s2, d2, b2, 128, 64, 0, hA, hB, as_, ad_, mm, dn, el, ei, stream);

    float* out = (float*)d_out;
    fill_f32<<<(N_GRAPHS * OUT_C + 255) / 256, 256, 0, stream>>>(out, 0.f, N_GRAPHS * OUT_C);
    pool_kernel<<<(N_NODES * OUT_C + 255) / 256, 256, 0, stream>>>(hB, batch, out);
}